// STGNN_50800873177459
// MI455X (gfx1250) — compile-verified
//
#include <hip/hip_runtime.h>
#include <hip/hip_bf16.h>

// ---------------------------------------------------------------------------
// STGNN on MI455X (gfx1250, wave32).
// Key algebraic reduction: only the last 15 timesteps feed the output
// (dilated causal conv receptive field), so only 120 GCN instances are
// needed. Dominant GEMMs (AN[1024x1024] @ Y) run on V_WMMA_F32_16X16X4_F32
// with fp32 accumulation for exact-precision matrix math.
// ---------------------------------------------------------------------------

#define B_ 8
#define S_ 512
#define N_ 1024
#define F_ 32
#define H_ 64
#define G0_ 32
#define SL_ 15            // receptive field of last output
#define S0_ (S_ - SL_)    // 497
#define INST_ (B_ * SL_)  // 120

typedef float v8f __attribute__((ext_vector_type(8)));
typedef float v2f __attribute__((ext_vector_type(2)));

// ---------------- workspace layout (in floats) ----------------
#define OFF_DIS 0u
#define OFF_AN  1024u
#define OFF_Y0  (OFF_AN + (unsigned)(N_ * N_))
#define OFF_H1  (OFF_Y0 + (unsigned)(INST_ * N_ * G0_))
#define OFF_Y1  (OFF_H1 + (unsigned)(INST_ * N_ * G0_))
#define OFF_SP  (OFF_Y1 + (unsigned)(INST_ * N_ * H_))

// ---------------------------------------------------------------------------
// 1) dis[m] = rsqrt(sum_n adj[m][n] + 1)
// ---------------------------------------------------------------------------
__global__ void k_dis(const float* __restrict__ adj, float* __restrict__ dis) {
  __shared__ float red[256];
  const int m = blockIdx.x;
  float s = 0.f;
  for (int n = threadIdx.x; n < N_; n += 256) s += adj[m * N_ + n];
  red[threadIdx.x] = s;
  __syncthreads();
  for (int off = 128; off > 0; off >>= 1) {
    if ((int)threadIdx.x < off) red[threadIdx.x] += red[threadIdx.x + off];
    __syncthreads();
  }
  if (threadIdx.x == 0) {
    float deg = red[0] + 1.0f;
    dis[m] = deg > 0.f ? rsqrtf(deg) : 0.f;
  }
}

// ---------------------------------------------------------------------------
// 2) an[m][n] = dis[m]*(adj[m][n] + (m==n))*dis[n]
// ---------------------------------------------------------------------------
__global__ void k_an(const float* __restrict__ adj, const float* __restrict__ dis,
                     float* __restrict__ an) {
  const int idx = blockIdx.x * 256 + threadIdx.x;
  const int m = idx >> 10;
  const int n = idx & (N_ - 1);
  float a = adj[idx] + (m == n ? 1.0f : 0.0f);
  an[idx] = dis[m] * a * dis[n];
}

// ---------------------------------------------------------------------------
// 3) zero spatial accumulator
// ---------------------------------------------------------------------------
__global__ void k_zero(float* __restrict__ p, int n) {
  int i = blockIdx.x * 256 + threadIdx.x;
  if (i < n) p[i] = 0.f;
}

// ---------------------------------------------------------------------------
// 4) Y0[inst][node][g] = sum_f x[b][s0+si][node][f] * w0[f][g]
// ---------------------------------------------------------------------------
__global__ void k_y0(const float* __restrict__ x, const float* __restrict__ w0,
                     float* __restrict__ y0) {
  __shared__ float Ws[F_ * G0_];
  __shared__ float Xs[8][F_];
  const int inst = blockIdx.y;
  const int b = inst / SL_;
  const int s = S0_ + (inst % SL_);
  const int t = threadIdx.x;
  for (int i = t; i < F_ * G0_; i += 256) Ws[i] = w0[i];
  {
    const int nl = t >> 5, f = t & 31;
    const int node = blockIdx.x * 8 + nl;
    Xs[nl][f] = x[(((size_t)b * S_ + s) * N_ + node) * F_ + f];
  }
  __syncthreads();
  const int nl = t >> 5, g = t & 31;
  const int node = blockIdx.x * 8 + nl;
  float acc = 0.f;
#pragma unroll
  for (int f = 0; f < F_; ++f) acc += Xs[nl][f] * Ws[f * G0_ + g];
  y0[((size_t)inst * N_ + node) * G0_ + g] = acc;
}

// ---------------------------------------------------------------------------
// 5/7) aggregate: Hout = relu(AN @ Yin + bias), optional node-mean into spatial.
//      block = 128 threads (4 waves), each wave owns 16 output rows.
//      A (64x64) and B (64xC) tiles staged in LDS; f32 16x16x4 WMMA.
// ---------------------------------------------------------------------------
template <int C, bool STORE, bool MEAN>
__global__ __launch_bounds__(128) void aggregate(
    const float* __restrict__ an, const float* __restrict__ yin,
    const float* __restrict__ bias, float* __restrict__ hout,
    float* __restrict__ spatial) {
  __shared__ float As[64][65];
  __shared__ float Bs[64][C + 1];

  const int inst = blockIdx.y;
  const int rowBlock = blockIdx.x * 64;
  const int wave = threadIdx.x >> 5;
  const int lane = threadIdx.x & 31;
  const int half = lane >> 4;        // 0 or 1
  const int l16 = lane & 15;
  const float* ybase = yin + (size_t)inst * N_ * C;

  constexpr int CT = C / 16;
  v8f acc[CT];
#pragma unroll
  for (int ct = 0; ct < CT; ++ct) acc[ct] = (v8f){0.f, 0.f, 0.f, 0.f, 0.f, 0.f, 0.f, 0.f};

  for (int kb = 0; kb < N_; kb += 64) {
    // cooperative load of A tile (64 rows x 64 k), coalesced along k
    for (int i = threadIdx.x; i < 64 * 64; i += 128) {
      int r = i >> 6, kk = i & 63;
      As[r][kk] = an[(size_t)(rowBlock + r) * N_ + kb + kk];
    }
    // cooperative load of B tile (64 k x C cols), coalesced along c
    for (int i = threadIdx.x; i < 64 * C; i += 128) {
      int kk = i / C, c = i % C;
      Bs[kk][c] = ybase[(size_t)(kb + kk) * C + c];
    }
    __syncthreads();

    const int rl = wave * 16 + l16;
#pragma unroll
    for (int ks = 0; ks < 64; ks += 4) {
      const int kf = ks + half * 2;
      v2f a;
      a.x = As[rl][kf];
      a.y = As[rl][kf + 1];
#pragma unroll
      for (int ct = 0; ct < CT; ++ct) {
        v2f bfrag;
        bfrag.x = Bs[kf][ct * 16 + l16];
        bfrag.y = Bs[kf + 1][ct * 16 + l16];
        acc[ct] = __builtin_amdgcn_wmma_f32_16x16x4_f32(
            false, a, false, bfrag, (short)0, acc[ct], false, false);
      }
    }
    __syncthreads();
  }

  // epilogue: bias + relu (+ store / + mean accumulate)
  const int R = rowBlock + wave * 16;
#pragma unroll
  for (int ct = 0; ct < CT; ++ct) {
    const int n = ct * 16 + l16;
    const float bn = bias[n];
    float msum = 0.f;
#pragma unroll
    for (int v = 0; v < 8; ++v) {
      const int m = R + half * 8 + v;
      float val = acc[ct][v] + bn;
      val = fmaxf(val, 0.f);
      if (STORE) hout[((size_t)inst * N_ + m) * C + n] = val;
      if (MEAN) msum += val;
    }
    if (MEAN) atomicAdd(&spatial[inst * H_ + n], msum * (1.0f / (float)N_));
  }
}

// ---------------------------------------------------------------------------
// 6) Y1[inst][node][h] = sum_g H1[inst][node][g] * w1[g][h]
// ---------------------------------------------------------------------------
__global__ void k_y1(const float* __restrict__ h1, const float* __restrict__ w1,
                     float* __restrict__ y1) {
  __shared__ float Ws[G0_ * H_];
  __shared__ float Hs[4][G0_];
  const int inst = blockIdx.y;
  const int t = threadIdx.x;
  for (int i = t; i < G0_ * H_; i += 256) Ws[i] = w1[i];
  if (t < 128) {
    const int nl = t >> 5, g = t & 31;
    const int node = blockIdx.x * 4 + nl;
    Hs[nl][g] = h1[((size_t)inst * N_ + node) * G0_ + g];
  }
  __syncthreads();
  const int nl = t >> 6, h = t & 63;
  const int node = blockIdx.x * 4 + nl;
  float acc = 0.f;
#pragma unroll
  for (int g = 0; g < G0_; ++g) acc += Hs[nl][g] * Ws[g * H_ + h];
  y1[((size_t)inst * N_ + node) * H_ + h] = acc;
}

// ---------------------------------------------------------------------------
// 8) temporal head: dilated causal convs on 15 steps + fusion + predictor.
//    one block per batch, 64 threads (one per channel).
// ---------------------------------------------------------------------------
__global__ void k_head(const float* __restrict__ spatial,
                       const float* __restrict__ cw0, const float* __restrict__ cb0,
                       const float* __restrict__ cw1, const float* __restrict__ cb1,
                       const float* __restrict__ cw2, const float* __restrict__ cb2,
                       const float* __restrict__ fw, const float* __restrict__ fb,
                       const float* __restrict__ pw1, const float* __restrict__ pb1,
                       const float* __restrict__ pw2, const float* __restrict__ pb2,
                       float* __restrict__ out) {
  __shared__ float t0s[SL_][H_];
  __shared__ float t1s[SL_][H_];
  __shared__ float t2s[SL_][H_];
  __shared__ float lastS[H_];
  __shared__ float fusedS[H_];
  __shared__ float p1S[H_ / 2];

  const int b = blockIdx.x;
  const int o = threadIdx.x;  // channel

  for (int ls = 0; ls < SL_; ++ls)
    t0s[ls][o] = spatial[((size_t)b * SL_ + ls) * H_ + o];
  __syncthreads();

  // conv1, dilation 1: outputs needed at ls = 2,4,...,14
  for (int ls = 2; ls <= 14; ls += 2) {
    float acc = cb0[o];
    for (int i = 0; i < H_; ++i) {
      const float* w = &cw0[(o * H_ + i) * 3];
      acc += w[0] * t0s[ls - 2][i] + w[1] * t0s[ls - 1][i] + w[2] * t0s[ls][i];
    }
    t1s[ls][o] = fmaxf(acc, 0.f);
  }
  __syncthreads();

  // conv2, dilation 2: outputs needed at ls = 6,10,14
  for (int ls = 6; ls <= 14; ls += 4) {
    float acc = cb1[o];
    for (int i = 0; i < H_; ++i) {
      const float* w = &cw1[(o * H_ + i) * 3];
      acc += w[0] * t1s[ls - 4][i] + w[1] * t1s[ls - 2][i] + w[2] * t1s[ls][i];
    }
    t2s[ls][o] = fmaxf(acc, 0.f);
  }
  __syncthreads();

  // conv3, dilation 4: only last output (ls = 14)
  {
    float acc = cb2[o];
    for (int i = 0; i < H_; ++i) {
      const float* w = &cw2[(o * H_ + i) * 3];
      acc += w[0] * t2s[6][i] + w[1] * t2s[10][i] + w[2] * t2s[14][i];
    }
    lastS[o] = fmaxf(acc, 0.f);
  }
  __syncthreads();

  // fusion: relu(last @ fusion_w + fb)
  {
    float acc = fb[o];
    for (int i = 0; i < H_; ++i) acc += lastS[i] * fw[i * H_ + o];
    fusedS[o] = fmaxf(acc, 0.f);
  }
  __syncthreads();

  // pred layer 1: relu(fused @ pred_w1 + pb1), width 32
  if (o < H_ / 2) {
    float acc = pb1[o];
    for (int i = 0; i < H_; ++i) acc += fusedS[i] * pw1[i * (H_ / 2) + o];
    p1S[o] = fmaxf(acc, 0.f);
  }
  __syncthreads();

  if (o == 0) {
    float acc = pb2[0];
    for (int j = 0; j < H_ / 2; ++j) acc += p1S[j] * pw2[j];
    out[b] = acc;
  }
}

// ---------------------------------------------------------------------------
extern "C" void kernel_launch(void* const* d_in, const int* in_sizes, int n_in,
                              void* d_out, int out_size, void* d_ws, size_t ws_size,
                              hipStream_t stream) {
  const float* x    = (const float*)d_in[0];
  const float* adj  = (const float*)d_in[1];
  const float* w0   = (const float*)d_in[2];
  const float* b0   = (const float*)d_in[3];
  const float* w1   = (const float*)d_in[4];
  const float* b1   = (const float*)d_in[5];
  const float* cw0  = (const float*)d_in[6];
  const float* cb0  = (const float*)d_in[7];
  const float* cw1  = (const float*)d_in[8];
  const float* cb1  = (const float*)d_in[9];
  const float* cw2  = (const float*)d_in[10];
  const float* cb2  = (const float*)d_in[11];
  const float* fw   = (const float*)d_in[12];
  const float* fb   = (const float*)d_in[13];
  const float* pw1  = (const float*)d_in[14];
  const float* pb1  = (const float*)d_in[15];
  const float* pw2  = (const float*)d_in[16];
  const float* pb2  = (const float*)d_in[17];
  float* out = (float*)d_out;

  float* ws = (float*)d_ws;
  float* dis = ws + OFF_DIS;
  float* an  = ws + OFF_AN;
  float* y0  = ws + OFF_Y0;
  float* h1  = ws + OFF_H1;
  float* y1  = ws + OFF_Y1;
  float* sp  = ws + OFF_SP;

  // 1) degree rsqrt
  k_dis<<<N_, 256, 0, stream>>>(adj, dis);
  // 2) normalized adjacency
  k_an<<<(N_ * N_) / 256, 256, 0, stream>>>(adj, dis, an);
  // 3) zero spatial accumulator
  k_zero<<<(INST_ * H_ + 255) / 256, 256, 0, stream>>>(sp, INST_ * H_);
  // 4) Y0 = X @ W0  (last 15 timesteps only)
  k_y0<<<dim3(N_ / 8, INST_), 256, 0, stream>>>(x, w0, y0);
  // 5) H1 = relu(AN @ Y0 + b0)   [WMMA]
  aggregate<G0_, true, false><<<dim3(N_ / 64, INST_), 128, 0, stream>>>(
      an, y0, b0, h1, nullptr);
  // 6) Y1 = H1 @ W1
  k_y1<<<dim3(N_ / 4, INST_), 256, 0, stream>>>(h1, w1, y1);
  // 7) spatial = mean_nodes(relu(AN @ Y1 + b1))   [WMMA + fused mean]
  aggregate<H_, false, true><<<dim3(N_ / 64, INST_), 128, 0, stream>>>(
      an, y1, b1, nullptr, sp);
  // 8) temporal convs + head
  k_head<<<B_, H_, 0, stream>>>(sp, cw0, cb0, cw1, cb1, cw2, cb2,
                                fw, fb, pw1, pb1, pw2, pb2, out);
}